// EquidistantDiscreteContinuousConv3d_87170656239871
// MI455X (gfx1250) — compile-verified
//
#include <hip/hip_runtime.h>

// MI455X / gfx1250 implicit-GEMM 3D conv using V_WMMA_F32_16X16X4_F32 with
// async global->LDS staging (GLOBAL_LOAD_ASYNC_TO_LDS_B32 / ASYNCcnt).
// out(2,16,64,64,64) = conv3d(x(2,8,128,128,128), fold(weight,psi)(16,8,5,5,5),
//                             stride 2, pad 2) + bias

typedef float v2f __attribute__((ext_vector_type(2)));
typedef float v8f __attribute__((ext_vector_type(8)));

#define IN_D   128
#define OUT_D  64
#define NC     8        // input channels
#define OC     16       // output channels
#define KS     33       // basis functions
#define TAPS   125      // 5*5*5
#define XPAD   68       // padded x extent of the LDS tile (67 used + 1 pad)
#define PPITCH (25*XPAD)          // float2 pairs per channel-pair plane-set
#define TILE_PAIRS (4*PPITCH)     // 6800 float2 = 54.4 KB LDS
#define X_CH_BYTES (IN_D*IN_D*IN_D*4)  // byte stride between channels in x

// ---------------------------------------------------------------------------
// Prep: fold weight (O,C,33) with psi (33,125) into the dense conv kernel,
// stored directly in WMMA-B operand lane layout:
//   Bmat[kc][lane][v] = Wk[4*kc + 2*(lane>>4) + v][o = lane&15]
// with K ordered as k = tap*8 + c  (so a K-chunk of 4 = 4 consecutive c's).
// Total 250*32*2 = 16000 floats = 64 KB in workspace.
// ---------------------------------------------------------------------------
__global__ void disco_fold_kernel(const float* __restrict__ weight,
                                  const float* __restrict__ psi,
                                  float* __restrict__ Bmat) {
  int t = blockIdx.x * blockDim.x + threadIdx.x;
  if (t >= 250 * 64) return;
  int kc   = t >> 6;          // 0..249
  int r    = t & 63;
  int lane = r >> 1;
  int v    = r & 1;
  int hi   = lane >> 4;
  int o    = lane & 15;
  int tap  = kc >> 1;                          // 0..124
  int c    = ((kc & 1) << 2) + (hi << 1) + v;  // 0..7
  float acc = 0.f;
  const float* w = weight + (o * NC + c) * KS;
  for (int j = 0; j < KS; ++j) acc = fmaf(w[j], psi[j * TAPS + tap], acc);
  Bmat[t] = acc;
}

// ---------------------------------------------------------------------------
// Conv: one block = 64 threads (2 waves). Block handles (n, zo, yo, x-half):
// 32 xo outputs x all 16 output channels. Each wave does a 16x16 WMMA tile
// (M = 16 consecutive xo, N = 16 output channels), K = 1000 in chunks of 4.
//
// LDS tile is channel-PAIR interleaved: sx2[(cp*25 + dz*5+dy)*XPAD + xp] is a
// float2 holding channels {2cp, 2cp+1}. Each WMMA A-fragment is then exactly
// one aligned ds_load_b64 straight into an even VGPR pair.
//
// Staging uses GLOBAL_LOAD_ASYNC_TO_LDS_B32 (GVS mode: SGPR base + i32 byte
// offset) for in-range cells; zero-padding cells get a ds_store_b64 to
// disjoint LDS addresses (no ds<->async hazard). s_wait_asynccnt 0 +
// workgroup barrier close the staging pipeline.
// ---------------------------------------------------------------------------
__global__ __launch_bounds__(64)
void disco_conv_kernel(const float* __restrict__ x,
                       const float* __restrict__ Bmat,
                       const float* __restrict__ bias,
                       float* __restrict__ out) {
  __shared__ float2 sx2[TILE_PAIRS];

  int b  = blockIdx.x;
  int xh = b & 1;              // x half: xo in [32*xh, 32*xh+31]
  int yo = (b >> 1) & 63;
  int zo = (b >> 7) & 63;
  int n  = b >> 13;
  int tid = threadIdx.x;

  // Hint the weight matrix into cache while we stage the input tile.
  if (tid < 32) __builtin_prefetch(Bmat + tid * 500, 0, 1);

  // ---- async-stage input halo tile into LDS (zero OOB -> padding) ----
  unsigned sbase = (unsigned)(uintptr_t)(&sx2[0]);   // flat[31:0] == LDS offset
  unsigned long long xbase_sgpr = (unsigned long long)(uintptr_t)x;
  for (int p = tid; p < TILE_PAIRS; p += 64) {
    int cp  = p / PPITCH;            // channel pair 0..3
    int rem = p - cp * PPITCH;
    int zy  = rem / XPAD;
    int xp  = rem - zy * XPAD;
    int dz  = zy / 5, dy = zy - dz * 5;
    int zi  = 2 * zo + dz - 2;
    int yi  = 2 * yo + dy - 2;
    int xi  = 64 * xh + xp - 2;
    unsigned laddr = sbase + 8u * (unsigned)p;
    if ((unsigned)zi < 128u && (unsigned)yi < 128u && (unsigned)xi < 128u && xp < 67) {
      int goff = 4 * ((((n * NC + 2 * cp) * IN_D + zi) * IN_D + yi) * IN_D + xi);
      unsigned laddr1 = laddr + 4u;
      int goff1 = goff + X_CH_BYTES;     // next channel of the pair
      asm volatile("global_load_async_to_lds_b32 %0, %1, %2"
                   :: "v"(laddr), "v"(goff), "s"(xbase_sgpr) : "memory");
      asm volatile("global_load_async_to_lds_b32 %0, %1, %2"
                   :: "v"(laddr1), "v"(goff1), "s"(xbase_sgpr) : "memory");
    } else {
      float2 z = make_float2(0.f, 0.f);  // disjoint from async targets
      sx2[p] = z;
    }
  }
  asm volatile("s_wait_asynccnt 0x0" ::: "memory");
  __syncthreads();

  int lane = tid & 31;
  int wave = tid >> 5;
  int hi   = lane >> 4;        // lane half selects K rows {2hi, 2hi+1} of chunk
  int m    = lane & 15;        // M row (xo offset within wave tile)
  int xbase = 2 * (wave * 16 + m);   // x_pad column for this lane's output

  v8f acc = {};                      // 16x16 f32 accumulator (8 VGPRs/lane)
  const v2f* Bp = (const v2f*)Bmat;

  int kc = 0;
  for (int dz = 0; dz < 5; ++dz)
    for (int dy = 0; dy < 5; ++dy) {
      int rowoff = (dz * 5 + dy) * XPAD + xbase;
#pragma unroll
      for (int dx = 0; dx < 5; ++dx) {
#pragma unroll
        for (int cb = 0; cb < 2; ++cb) {
          // A fragment: channels {c, c+1}, c = cb*4 + 2*hi -> pair cp = cb*2+hi
          int aidx = (cb * 2 + hi) * PPITCH + rowoff + dx;
          v2f a = *(const v2f*)&sx2[aidx];          // one ds_load_b64
          v2f bv = Bp[kc * 32 + lane];              // one global_load_b64
          acc = __builtin_amdgcn_wmma_f32_16x16x4_f32(
              /*neg_a=*/false, a, /*neg_b=*/false, bv,
              /*c_mod=*/(short)0, acc, /*reuse_a=*/false, /*reuse_b=*/false);
          ++kc;
        }
      }
    }

  // ---- epilogue: D[M][N] -> lane holds N=o=m, VGPR r -> M = r + 8*hi ----
  int o = m;
  float bb = bias[o];
  int xo = xh * 32 + wave * 16 + 8 * hi;
  size_t obase = (((size_t)(n * OC + o) * OUT_D + zo) * OUT_D + yo) * OUT_D + xo;
  float4 lo = { acc[0] + bb, acc[1] + bb, acc[2] + bb, acc[3] + bb };
  float4 hv = { acc[4] + bb, acc[5] + bb, acc[6] + bb, acc[7] + bb };
  *(float4*)(out + obase)     = lo;   // global_store_b128, contiguous per lane
  *(float4*)(out + obase + 4) = hv;
}

// ---------------------------------------------------------------------------
extern "C" void kernel_launch(void* const* d_in, const int* in_sizes, int n_in,
                              void* d_out, int out_size, void* d_ws, size_t ws_size,
                              hipStream_t stream) {
  const float* x      = (const float*)d_in[0];  // (2,8,128,128,128)
  const float* weight = (const float*)d_in[1];  // (16,8,33)
  const float* bias   = (const float*)d_in[2];  // (16,)
  const float* psi    = (const float*)d_in[3];  // (33,5,5,5)
  float* Bmat = (float*)d_ws;                   // 16000 floats = 64 KB scratch

  disco_fold_kernel<<<(250 * 64 + 255) / 256, 256, 0, stream>>>(weight, psi, Bmat);

  // grid: n(2) * zo(64) * yo(64) * xhalf(2) = 16384 blocks, 64 threads each
  disco_conv_kernel<<<2 * 64 * 64 * 2, 64, 0, stream>>>(
      x, Bmat, bias, (float*)d_out);
}